// Easy_SS2D_29832842838266
// MI455X (gfx1250) — compile-verified
//
#include <hip/hip_runtime.h>
#include <hip/hip_bf16.h>
#include <math.h>

// ---- problem constants (from reference) ----
#define BSZ   4
#define DM    96
#define DI    192
#define HH    128
#define WW    128
#define LL    (HH*WW)        // 16384
#define KDIR  4
#define DTR   6
#define NSEQ  (BSZ*KDIR*DI)  // 3072 scan sequences
#define CHUNK 512
#define NCHUNK (LL/CHUNK)    // 32 chunks per sequence

typedef __attribute__((ext_vector_type(2))) float v2f;
typedef __attribute__((ext_vector_type(8))) float v8f;

__device__ __forceinline__ float sigmoidf_(float x){ return 1.0f/(1.0f+__expf(-x)); }
__device__ __forceinline__ float softplusf_(float x){ return (x>20.0f)?x:log1pf(__expf(x)); }

// scan-order index l -> original pixel index p, per direction k
__device__ __forceinline__ int perm_k(int k, int l){
  switch(k&3){
    case 0:  return l;                                   // row-major
    case 1:  return ((l&(HH-1))*WW) + (l>>7);            // transposed
    case 2:  return LL-1-l;                              // reversed
    default: { int m=LL-1-l; return ((m&(HH-1))*WW)+(m>>7); } // transposed+reversed
  }
}

// ---------------------------------------------------------------------------
// Generic fp32 WMMA GEMM:  Y[b][m][p] = sum_k A[m][k] * X[b][k][p] + bias[m]
// One wave computes one 16x16 output tile via V_WMMA_F32_16X16X4_F32.
// VGPR layouts per CDNA5 ISA 7.12.2:
//   A 16x4 : lanes 0-15 hold {K0,K1} in {v0,v1}, lanes 16-31 hold {K2,K3}
//   B 4x16 : v0 = rows K0(K1 in hi lanes), v1 = rows K2/K3, N = lane%16
//   C/D    : vgpr r -> M = r (+8 for hi lanes), N = lane%16
// ---------------------------------------------------------------------------
__global__ __launch_bounds__(256) void wmma_gemm_f32(
    const float* __restrict__ A,    // [M][Kd] row-major
    const float* __restrict__ X,    // [BSZ][Kd][LL]
    const float* __restrict__ bias, // [M]
    float* __restrict__ Y,          // [BSZ][M][LL]
    int M, int Kd)
{
  const int lane = threadIdx.x & 31;
  const int wid  = threadIdx.x >> 5;
  const int mtiles = M >> 4;
  const int tile = blockIdx.x * 8 + wid;
  const int mt = tile % mtiles;
  const int nt = tile / mtiles;
  const int n0 = nt << 4;
  const int bb = n0 / LL;             // 16-wide N tile never crosses a batch
  const int p0 = n0 % LL;
  const int lm = lane & 15;
  const int hi = lane >> 4;
  const int m0 = mt << 4;

  const float* Xb = X + (size_t)bb * Kd * LL + p0 + lm;
  const float* Ar = A + (size_t)(m0 + lm) * Kd + 2*hi;

  v8f c = {0.f,0.f,0.f,0.f,0.f,0.f,0.f,0.f};
  for (int kb = 0; kb < Kd; kb += 4) {
    v2f a, b;
    a.x = Ar[kb];
    a.y = Ar[kb + 1];
    b.x = Xb[(size_t)(kb + hi) * LL];
    b.y = Xb[(size_t)(kb + 2 + hi) * LL];
    c = __builtin_amdgcn_wmma_f32_16x16x4_f32(false, a, false, b,
                                              (short)0, c, false, false);
  }
  float* Yb = Y + (size_t)bb * M * LL;
  #pragma unroll
  for (int r = 0; r < 8; ++r) {
    int row = m0 + r + 8*hi;
    Yb[(size_t)row * LL + p0 + lm] = c[r] + bias[row];
  }
}

// ---------------------------------------------------------------------------
// Depthwise 3x3 conv (SAME, zero pad) + bias + SiLU
// ---------------------------------------------------------------------------
__global__ __launch_bounds__(256) void dwconv_silu(
    const float* __restrict__ z, const float* __restrict__ w,
    const float* __restrict__ b, float* __restrict__ u)
{
  int idx = blockIdx.x*blockDim.x + threadIdx.x;
  if (idx >= BSZ*DI*LL) return;
  int p = idx % LL; int d = (idx/LL) % DI;
  int h = p >> 7, wc = p & (WW-1);
  const float* zp = z + (size_t)(idx - p);   // start of this (b,d) plane
  const float* wk = w + d*9;
  float s = b[d];
  #pragma unroll
  for (int i=-1;i<=1;++i){
    int h2=h+i; if (h2<0||h2>=HH) continue;
    #pragma unroll
    for (int j=-1;j<=1;++j){
      int w2=wc+j; if (w2<0||w2>=WW) continue;
      s += wk[(i+1)*3+(j+1)] * zp[h2*WW + w2];
    }
  }
  u[idx] = s * sigmoidf_(s);
}

// ---------------------------------------------------------------------------
// Per-pixel low-rank projections: xd[b][k*8+e][p] = sum_d W[k][e][d]*u[b][d][p]
// (projection commutes with the scan permutations, so do it in pixel order)
// ---------------------------------------------------------------------------
__global__ __launch_bounds__(256) void xproj(
    const float* __restrict__ u, const float* __restrict__ W,
    float* __restrict__ xd)
{
  int idx = blockIdx.x*blockDim.x + threadIdx.x;   // (b,p)
  if (idx >= BSZ*LL) return;
  int p = idx % LL; int bi = idx / LL;
  const float* ub = u + (size_t)bi*DI*LL + p;
  float acc[KDIR*8];
  #pragma unroll
  for (int e=0;e<KDIR*8;++e) acc[e]=0.f;
  for (int d=0; d<DI; ++d){
    float uv = ub[(size_t)d*LL];
    #pragma unroll
    for (int e=0;e<KDIR*8;++e) acc[e] += W[(size_t)e*DI + d] * uv;
  }
  float* xo = xd + (size_t)bi*KDIR*8*LL + p;
  #pragma unroll
  for (int e=0;e<KDIR*8;++e) xo[(size_t)e*LL] = acc[e];
}

// ---------------------------------------------------------------------------
// Chunked selective scan, d = fastest-varying lane index so all xd loads are
// wave-uniform (one cacheline broadcast per step instead of 32).
// final_pass=0: per-chunk aggregates (prodA, acc).
// final_pass=1: replay with carried-in state, emit y at inverse-permuted pixel
// so the 4-direction merge becomes a plain sum.
// Launch with blockDim = DI (192 = 6 waves) so (b,k,ch) is wave-uniform.
// ---------------------------------------------------------------------------
__global__ __launch_bounds__(192) void scan_pass(
    const float* __restrict__ u,   const float* __restrict__ xd,
    const float* __restrict__ dtw, const float* __restrict__ dtb,
    const float* __restrict__ A_logs, const float* __restrict__ Ds,
    float* __restrict__ aggA, float* __restrict__ aggB,
    const float* __restrict__ carryIn, float* __restrict__ yall,
    int final_pass)
{
  int idx = blockIdx.x*blockDim.x + threadIdx.x;
  if (idx >= NSEQ*NCHUNK) return;
  const int d  = idx % DI;                 // fastest across lanes
  const int t  = idx / DI;
  const int ch = t % NCHUNK;
  const int k  = (t / NCHUNK) % KDIR;
  const int bi = t / (NCHUNK * KDIR);
  const int seq = (bi*KDIR + k)*DI + d;
  const int aix = seq*NCHUNK + ch;

  const float* xb = xd + ((size_t)bi*KDIR + k)*8*LL;   // wave-uniform base
  const float* ub = u  + ((size_t)bi*DI  + d)*LL;

  float wr[DTR];
  #pragma unroll
  for (int r=0;r<DTR;++r) wr[r] = dtw[((size_t)k*DI + d)*DTR + r];
  const float bias = dtb[k*DI + d];
  const float Av   = -__expf(A_logs[k*DI + d]);
  const float Dv   = Ds[k*DI + d];

  float h = final_pass ? carryIn[aix] : 0.f;
  float prod = 1.f;
  const int l0 = ch * CHUNK;
  for (int l = l0; l < l0 + CHUNK; ++l){
    int p = perm_k(k, l);
    {  // prefetch the per-thread streaming u walk ~16 steps ahead
      int lp = l + 16; if (lp >= LL) lp = LL-1;
      __builtin_prefetch(ub + perm_k(k, lp), 0, 3);
    }
    float dt = bias;
    #pragma unroll
    for (int r=0;r<DTR;++r) dt += wr[r] * xb[(size_t)r*LL + p];
    float delta = softplusf_(dt);
    float dA  = __expf(delta * Av);
    float uv  = ub[p];
    float dBu = delta * xb[(size_t)6*LL + p] * uv;
    h = h*dA + dBu;
    if (final_pass) {
      yall[(size_t)seq*LL + p] = h * xb[(size_t)7*LL + p] + Dv*uv;
    } else {
      prod *= dA;
    }
  }
  if (!final_pass){ aggA[aix] = prod; aggB[aix] = h; }
}

// inter-chunk sequential prefix (32 steps per sequence, 3072 threads)
__global__ __launch_bounds__(256) void chunk_prefix(
    const float* __restrict__ aggA, const float* __restrict__ aggB,
    float* __restrict__ carry)
{
  int seq = blockIdx.x*blockDim.x + threadIdx.x;
  if (seq >= NSEQ) return;
  float c = 0.f;
  for (int ch=0; ch<NCHUNK; ++ch){
    carry[seq*NCHUNK + ch] = c;
    c = c*aggA[seq*NCHUNK + ch] + aggB[seq*NCHUNK + ch];
  }
}

// ---------------------------------------------------------------------------
// Merge 4 directions + channel LayerNorm (population var, eps=1e-5)
// ---------------------------------------------------------------------------
__global__ __launch_bounds__(256) void merge_ln(
    const float* __restrict__ yall, const float* __restrict__ lnw,
    const float* __restrict__ lnb,  float* __restrict__ yn)
{
  int idx = blockIdx.x*blockDim.x + threadIdx.x;   // (b,p)
  if (idx >= BSZ*LL) return;
  int p = idx % LL; int bi = idx/LL;
  const float* yb = yall + (size_t)bi*KDIR*DI*LL + p;
  float* yo = yn + (size_t)bi*DI*LL + p;
  float s1=0.f, s2=0.f;
  for (int d=0; d<DI; ++d){
    float v = yb[(size_t)d*LL] + yb[((size_t)DI+d)*LL]
            + yb[((size_t)2*DI+d)*LL] + yb[((size_t)3*DI+d)*LL];
    s1 += v; s2 += v*v;
    yo[(size_t)d*LL] = v;
  }
  float mu  = s1 * (1.f/DI);
  float var = s2 * (1.f/DI) - mu*mu;
  float rs  = rsqrtf(var + 1e-5f);
  for (int d=0; d<DI; ++d){
    float v = yo[(size_t)d*LL];
    yo[(size_t)d*LL] = (v - mu)*rs*lnw[d] + lnb[d];
  }
}

// ---------------------------------------------------------------------------
extern "C" void kernel_launch(void* const* d_in, const int* in_sizes, int n_in,
                              void* d_out, int out_size, void* d_ws, size_t ws_size,
                              hipStream_t stream)
{
  const float* x      = (const float*)d_in[0];
  const float* in_w   = (const float*)d_in[1];
  const float* in_b   = (const float*)d_in[2];
  const float* dw_w   = (const float*)d_in[3];
  const float* dw_b   = (const float*)d_in[4];
  const float* xpw    = (const float*)d_in[5];
  const float* dtw    = (const float*)d_in[6];
  const float* dtb    = (const float*)d_in[7];
  const float* A_logs = (const float*)d_in[8];
  const float* Ds     = (const float*)d_in[9];
  const float* ln_w   = (const float*)d_in[10];
  const float* ln_b   = (const float*)d_in[11];
  const float* out_w  = (const float*)d_in[12];
  const float* out_b  = (const float*)d_in[13];

  // workspace layout (floats); z buffer is reused as yn after the conv
  float* ws    = (float*)d_ws;
  float* z     = ws;                                  // BSZ*DI*LL
  float* u     = z    + (size_t)BSZ*DI*LL;            // BSZ*DI*LL
  float* xd    = u    + (size_t)BSZ*DI*LL;            // BSZ*KDIR*8*LL
  float* aggA  = xd   + (size_t)BSZ*KDIR*8*LL;        // NSEQ*NCHUNK
  float* aggB  = aggA + (size_t)NSEQ*NCHUNK;          // NSEQ*NCHUNK
  float* carry = aggB + (size_t)NSEQ*NCHUNK;          // NSEQ*NCHUNK
  float* yall  = carry+ (size_t)NSEQ*NCHUNK;          // BSZ*KDIR*DI*LL
  float* yn    = z;                                   // reuse (z dead after conv)

  // 1) in-proj GEMM: z[b][192][p] = in_w(192x96) @ x + in_b
  {
    int tiles = (DI/16) * (BSZ*LL/16);                // 12 * 4096
    wmma_gemm_f32<<<tiles/8, 256, 0, stream>>>(in_w, x, in_b, z, DI, DM);
  }
  // 2) depthwise conv + SiLU -> u
  dwconv_silu<<<(BSZ*DI*LL+255)/256, 256, 0, stream>>>(z, dw_w, dw_b, u);
  // 3a) per-pixel projections (dt_rank + B + C for all 4 directions)
  xproj<<<(BSZ*LL+255)/256, 256, 0, stream>>>(u, xpw, xd);
  // 3b) chunk aggregates (d-fastest mapping; block = DI = 6 waves)
  scan_pass<<<(NSEQ*NCHUNK)/DI, DI, 0, stream>>>(
      u, xd, dtw, dtb, A_logs, Ds, aggA, aggB, nullptr, nullptr, 0);
  // 3c) inter-chunk prefix
  chunk_prefix<<<(NSEQ+255)/256, 256, 0, stream>>>(aggA, aggB, carry);
  // 3d) replay with carry, emit y in original pixel order (merge-ready)
  scan_pass<<<(NSEQ*NCHUNK)/DI, DI, 0, stream>>>(
      u, xd, dtw, dtb, A_logs, Ds, nullptr, nullptr, carry, yall, 1);
  // 4a) merge directions + LayerNorm -> yn
  merge_ln<<<(BSZ*LL+255)/256, 256, 0, stream>>>(yall, ln_w, ln_b, yn);
  // 4b) out-proj GEMM: out[b][96][p] = out_w(96x192) @ yn + out_b
  {
    int tiles = (DM/16) * (BSZ*LL/16);                // 6 * 4096
    wmma_gemm_f32<<<tiles/8, 256, 0, stream>>>(out_w, yn, out_b,
                                               (float*)d_out, DM, DI);
  }
}